// SlotAttention_73332271612043
// MI455X (gfx1250) — compile-verified
//
#include <hip/hip_runtime.h>
#include <math.h>

// Problem constants (match reference): B=8, L=256, D=256, K=64
constexpr int Bb = 8;
constexpr int Ll = 256;
constexpr int Dd = 256;
constexpr int Kk = 64;

typedef __attribute__((ext_vector_type(2))) float v2f;
typedef __attribute__((ext_vector_type(8))) float v8f;

// D(16x16,f32) = A(16x4,f32) x B(4x16,f32) + C ; wave32, EXEC all-ones.
__device__ __forceinline__ v8f wmma_f32_4(v2f a, v2f b, v8f c) {
  // 8 args: (neg_a, A, neg_b, B, c_mod, C, reuse_a, reuse_b)
  return __builtin_amdgcn_wmma_f32_16x16x4_f32(false, a, false, b, (short)0, c,
                                               false, false);
}

// Hardware tanh (single trans-VALU op on CDNA5) instead of libm's ~20-op
// __ocml_tanh_f32 expansion. This sits in the dominant 33.5M-element loop.
__device__ __forceinline__ float fast_tanh(float x) {
#if __has_builtin(__builtin_amdgcn_tanhf)
  return __builtin_amdgcn_tanhf(x);
#else
  float r;
  asm volatile("v_tanh_f32 %0, %1" : "=v"(r) : "v"(x));
  return r;
#endif
}

// ---------------------------------------------------------------------------
// Kernel A: qi = x @ W1[:, :D]^T ; kj = x @ W1[:, D:]^T
// grid = B*(L/16) blocks, 256 threads (8 waves).
// Waves 0..3: qi tile columns n0 = wave*16 ; waves 4..7: kj tiles.
// Each wave produces one 16x16 f32 tile, accumulating over D in steps of 4.
// ---------------------------------------------------------------------------
__global__ __launch_bounds__(256) void qk_proj_kernel(
    const float* __restrict__ inp, const float* __restrict__ W1,
    float* __restrict__ qi, float* __restrict__ kj) {
  const int blk  = blockIdx.x;
  const int b    = blk / (Ll / 16);
  const int i0   = (blk % (Ll / 16)) * 16;
  const int wave = threadIdx.x >> 5;
  const int lane = threadIdx.x & 31;
  const bool is_k = (wave >= 4);
  const int n0     = (wave & 3) * 16;
  const int coloff = is_k ? Dd : 0;

  // ISA f32 A-layout 16x4: lane l -> M = l&15 ; VGPR0/1 hold K = kh, kh+1
  // where kh = 0 (lanes 0-15) or 2 (lanes 16-31). B-layout mirrors (N = l&15).
  const int mn = lane & 15;
  const int kh = (lane >> 4) << 1;

  const float* arow = inp + ((size_t)b * Ll + (i0 + mn)) * Dd;      // x row
  const float* brow = W1 + (size_t)(n0 + mn) * (2 * Dd) + coloff;   // W1 row

  v8f acc = {};
#pragma unroll 4
  for (int d = 0; d < Dd; d += 4) {
    v2f a, bv;
    a.x  = arow[d + kh];
    a.y  = arow[d + kh + 1];
    bv.x = brow[d + kh];
    bv.y = brow[d + kh + 1];
    acc = wmma_f32_4(a, bv, acc);
  }

  // C/D f32 layout: VGPR r -> M = r + 8*(lane>=16), N = lane&15
  float* op     = is_k ? kj : qi;
  const int mb  = (lane >> 4) * 8;
#pragma unroll
  for (int r = 0; r < 8; ++r) {
    op[((size_t)b * Ll + (i0 + mb + r)) * Kk + (n0 + mn)] = acc[r];
  }
}

// ---------------------------------------------------------------------------
// Kernel B: scores[j] = sum_k tanh(qi[b,i,k] + kj[b,j,k]) * w2[k]; softmax_j.
// grid = B*L blocks (one per (b,i)), 256 threads (thread t <-> column j=t).
// Dominant cost of the whole op: 33.5M tanh, routed to the hardware
// V_TANH_F32 trans pipe; kj rows fetched as float4 (global_load_b128).
// ---------------------------------------------------------------------------
__global__ __launch_bounds__(256) void attn_softmax_kernel(
    const float* __restrict__ qi, const float* __restrict__ kj,
    const float* __restrict__ W2, float* __restrict__ attn) {
  __shared__ float q_s[Kk];
  __shared__ float w_s[Kk];
  __shared__ float r_s[Ll];

  const int b = blockIdx.x / Ll;
  const int i = blockIdx.x % Ll;
  const int t = threadIdx.x;

  if (t < Kk) {
    q_s[t] = qi[((size_t)b * Ll + i) * Kk + t];
    w_s[t] = W2[t];
  }
  __syncthreads();

  const float4* kr4 =
      reinterpret_cast<const float4*>(kj + ((size_t)b * Ll + t) * Kk);
  float s = 0.0f;
#pragma unroll
  for (int k4 = 0; k4 < Kk / 4; ++k4) {
    const float4 kv = kr4[k4];
    const int k = 4 * k4;
    s += fast_tanh(q_s[k + 0] + kv.x) * w_s[k + 0];
    s += fast_tanh(q_s[k + 1] + kv.y) * w_s[k + 1];
    s += fast_tanh(q_s[k + 2] + kv.z) * w_s[k + 2];
    s += fast_tanh(q_s[k + 3] + kv.w) * w_s[k + 3];
  }

  // block max
  r_s[t] = s;
  __syncthreads();
  for (int st = 128; st > 0; st >>= 1) {
    if (t < st) r_s[t] = fmaxf(r_s[t], r_s[t + st]);
    __syncthreads();
  }
  const float mx = r_s[0];
  __syncthreads();

  const float e = __expf(s - mx);

  // block sum
  r_s[t] = e;
  __syncthreads();
  for (int st = 128; st > 0; st >>= 1) {
    if (t < st) r_s[t] += r_s[t + st];
    __syncthreads();
  }

  attn[((size_t)b * Ll + i) * Ll + t] = e * (1.0f / r_s[0]);
}

// ---------------------------------------------------------------------------
// Kernel C: context = attn @ x  (per batch, 256x256x256 GEMM via WMMA f32),
// fused with concat: out[..., :D] = x ; out[..., D:] = context.
// grid = B*16*16 waves / 8 waves per block = 256 blocks, 256 threads.
// ---------------------------------------------------------------------------
__global__ __launch_bounds__(256) void context_kernel(
    const float* __restrict__ attn, const float* __restrict__ inp,
    float* __restrict__ out) {
  const int wave = threadIdx.x >> 5;
  const int lane = threadIdx.x & 31;
  const int g    = blockIdx.x * 8 + wave;  // global wave/tile id: 0..2047
  const int b    = g >> 8;
  const int rem  = g & 255;
  const int i0   = (rem >> 4) * 16;  // context row tile
  const int d0   = (rem & 15) * 16;  // context col tile

  const int mn = lane & 15;
  const int kh = (lane >> 4) << 1;

  const float* arow = attn + ((size_t)b * Ll + (i0 + mn)) * Ll;

  v8f acc = {};
  for (int j = 0; j < Ll; j += 4) {
    v2f a, bv;
    a.x = arow[j + kh];
    a.y = arow[j + kh + 1];
    const float* bp = inp + ((size_t)b * Ll + (j + kh)) * Dd + (d0 + mn);
    bv.x = bp[0];
    bv.y = bp[Dd];
    acc = wmma_f32_4(a, bv, acc);
  }

  const int mb = (lane >> 4) * 8;
#pragma unroll
  for (int r = 0; r < 8; ++r) {
    const int row      = i0 + mb + r;
    const size_t obase = ((size_t)b * Ll + row) * (2 * Dd);
    // second half: context
    out[obase + Dd + d0 + mn] = acc[r];
    // first half: pass-through copy of inputs (fused concat)
    out[obase + d0 + mn] = inp[((size_t)b * Ll + row) * Dd + d0 + mn];
  }
}

// ---------------------------------------------------------------------------
extern "C" void kernel_launch(void* const* d_in, const int* in_sizes, int n_in,
                              void* d_out, int out_size, void* d_ws,
                              size_t ws_size, hipStream_t stream) {
  const float* inp = (const float*)d_in[0];  // (B, L, D)
  const float* W1  = (const float*)d_in[1];  // (K, 2D)
  const float* W2  = (const float*)d_in[2];  // (1, K)
  float* out = (float*)d_out;                // (B, L, 2D)

  float* qi   = (float*)d_ws;                       // B*L*K floats (512 KB)
  float* kj   = qi + (size_t)Bb * Ll * Kk;          // B*L*K floats (512 KB)
  float* attn = kj + (size_t)Bb * Ll * Kk;          // B*L*L floats (2 MB)

  qk_proj_kernel<<<Bb * (Ll / 16), 256, 0, stream>>>(inp, W1, qi, kj);
  attn_softmax_kernel<<<Bb * Ll, 256, 0, stream>>>(qi, kj, W2, attn);
  context_kernel<<<(Bb * (Ll / 16) * (Dd / 16)) / 8, 256, 0, stream>>>(attn, inp, out);
}